// GraphLearning_20461224198295
// MI455X (gfx1250) — compile-verified
//
#include <hip/hip_runtime.h>
#include <hip/hip_bf16.h>

typedef float v2f __attribute__((ext_vector_type(2)));
typedef float v8f __attribute__((ext_vector_type(8)));

#define NREG 128
#define NIN_ 256
#define NH_ 128
#define HEADS_ 3
#define BATCH_ 8
#define EDGES_ 16256   // N*(N-1)
#define ROWS_ (BATCH_*NREG)   // 1024

enum { ACT_NONE = 0, ACT_ELU = 1, ACT_RELU = 2 };

__device__ __forceinline__ float act_elu(float v) { return v > 0.f ? v : __expf(v) - 1.f; }

// ---------------------------------------------------------------------------
// Generic batched GEMM: C[b] = act(A[b] @ B[b] + bias), fp32 via
// V_WMMA_F32_16X16X4_F32. One wave computes a 64x16 output block (4 row tiles)
// so each B fragment feeds 4 independent WMMA chains. M % 64 == 0, K % 4 == 0,
// F % 16 == 0.
// ---------------------------------------------------------------------------
__global__ __launch_bounds__(32)
void gemm16_wmma(const float* __restrict__ A, const float* __restrict__ Bw,
                 const float* __restrict__ bias, float* __restrict__ C,
                 int K, int lda, int ldb, int ldc,
                 long sA, long sB, long sC, int act)
{
    const int lane = threadIdx.x;        // 0..31
    const int half = lane >> 4;          // 0 or 1
    const int l16  = lane & 15;
    const int n0 = blockIdx.x * 16;
    const int m0 = blockIdx.y * 64;
    A  += (long)blockIdx.z * sA;
    Bw += (long)blockIdx.z * sB;
    C  += (long)blockIdx.z * sC;

    // A frag (16x4 fp32): lanes 0-15 -> K=0,1 ; lanes 16-31 -> K=2,3 (row = lane%16)
    const float* Ar0 = A + (long)(m0 +  0 + l16) * lda + 2 * half;
    const float* Ar1 = A + (long)(m0 + 16 + l16) * lda + 2 * half;
    const float* Ar2 = A + (long)(m0 + 32 + l16) * lda + 2 * half;
    const float* Ar3 = A + (long)(m0 + 48 + l16) * lda + 2 * half;
    // B frag (4x16 fp32): lanes 0-15 -> rows K=0,1 ; lanes 16-31 -> rows K=2,3
    const float* Bcol = Bw + (long)(2 * half) * ldb + n0 + l16;

    v8f acc0 = {}, acc1 = {}, acc2 = {}, acc3 = {};
    for (int k = 0; k < K; k += 4) {
        v2f b;
        b.x = Bcol[(long)k * ldb];
        b.y = Bcol[(long)(k + 1) * ldb];
        v2f a0, a1, a2, a3;
        a0.x = Ar0[k]; a0.y = Ar0[k + 1];
        a1.x = Ar1[k]; a1.y = Ar1[k + 1];
        a2.x = Ar2[k]; a2.y = Ar2[k + 1];
        a3.x = Ar3[k]; a3.y = Ar3[k + 1];
        acc0 = __builtin_amdgcn_wmma_f32_16x16x4_f32(false, a0, false, b, (short)0, acc0, false, false);
        acc1 = __builtin_amdgcn_wmma_f32_16x16x4_f32(false, a1, false, b, (short)0, acc1, false, false);
        acc2 = __builtin_amdgcn_wmma_f32_16x16x4_f32(false, a2, false, b, (short)0, acc2, false, false);
        acc3 = __builtin_amdgcn_wmma_f32_16x16x4_f32(false, a3, false, b, (short)0, acc3, false, false);
    }

    const float bv = bias ? bias[n0 + l16] : 0.f;
    v8f accs[4] = { acc0, acc1, acc2, acc3 };
    #pragma unroll
    for (int t = 0; t < 4; ++t) {
        #pragma unroll
        for (int v = 0; v < 8; ++v) {
            int row = m0 + 16 * t + v + 8 * half;   // C: VGPR v, lane-half selects row+8
            float val = accs[t][v] + bv;
            if (act == ACT_ELU)       val = act_elu(val);
            else if (act == ACT_RELU) val = fmaxf(val, 0.f);
            C[(long)row * ldc + n0 + l16] = val;
        }
    }
}

// ---------------------------------------------------------------------------
// Training-mode batch-norm over M rows, per feature column (grid.x = F).
// ---------------------------------------------------------------------------
__global__ __launch_bounds__(256)
void bn_kernel(const float* __restrict__ G, float* __restrict__ out,
               const float* __restrict__ gamma, const float* __restrict__ beta,
               int M, int F)
{
    const int f = blockIdx.x;
    __shared__ float ssum[256], ssq[256];
    float s = 0.f, q = 0.f;
    for (int m = threadIdx.x; m < M; m += 256) {
        float v = G[(long)m * F + f]; s += v; q += v * v;
    }
    ssum[threadIdx.x] = s; ssq[threadIdx.x] = q;
    __syncthreads();
    for (int st = 128; st > 0; st >>= 1) {
        if (threadIdx.x < st) { ssum[threadIdx.x] += ssum[threadIdx.x + st];
                                ssq[threadIdx.x]  += ssq[threadIdx.x + st]; }
        __syncthreads();
    }
    const float mu  = ssum[0] / (float)M;
    const float var = ssq[0] / (float)M - mu * mu;
    const float sc  = rsqrtf(var + 1e-5f) * gamma[f];
    const float be  = beta[f];
    for (int m = threadIdx.x; m < M; m += 256)
        out[(long)m * F + f] = (G[(long)m * F + f] - mu) * sc + be;
}

// GAT attention scores: s1[i]=H[i]·a1, s2[i]=H[i]·a2.  grid = B*HEADS, block = 128.
__global__ __launch_bounds__(128)
void gat_scores(const float* __restrict__ Hbuf, const float* __restrict__ gat_a,
                float* __restrict__ Sbuf)
{
    const int bh = blockIdx.x;                 // b*3+hd (Hbuf contiguous in bh)
    const int hd = bh % HEADS_;
    const float* H  = Hbuf + (long)bh * NH_ * NREG;
    const float* a1 = gat_a + hd * (2 * NH_);
    const float* a2 = a1 + NH_;
    const int i = threadIdx.x;
    float s1 = 0.f, s2 = 0.f;
    for (int c = 0; c < NH_; ++c) { float h = H[i * NH_ + c]; s1 += h * a1[c]; s2 += h * a2[c]; }
    Sbuf[bh * 256 + i]       = s1;
    Sbuf[bh * 256 + 128 + i] = s2;
}

// Row softmax of leaky_relu(s1[i]+s2[j], 0.2).  grid = B*HEADS*N (row i), block = 128 (j).
__global__ __launch_bounds__(128)
void gat_att(const float* __restrict__ Sbuf, float* __restrict__ Att)
{
    const int bh = blockIdx.x >> 7;
    const int i  = blockIdx.x & 127;
    const int j  = threadIdx.x;
    float e = Sbuf[bh * 256 + i] + Sbuf[bh * 256 + 128 + j];
    e = e > 0.f ? e : 0.2f * e;
    __shared__ float red[128];
    red[j] = e; __syncthreads();
    for (int st = 64; st > 0; st >>= 1) { if (j < st) red[j] = fmaxf(red[j], red[j + st]); __syncthreads(); }
    const float mx = red[0]; __syncthreads();
    const float ex = __expf(e - mx);
    red[j] = ex; __syncthreads();
    for (int st = 64; st > 0; st >>= 1) { if (j < st) red[j] += red[j + st]; __syncthreads(); }
    Att[(long)bh * NREG * NREG + i * NREG + j] = ex / red[0];
}

// Per-edge gate: xr = elu( elu(Ps[b,j]+Pr[b,i]+b1) · w2 + b2 ).  One wave per (b,j,i).
// Accumulates global sum/sumsq over the E real edges per batch for f2's batch-norm.
__global__ __launch_bounds__(256)
void edge_gate(const float* __restrict__ Ps, const float* __restrict__ Pr,
               const float* __restrict__ b1v, const float* __restrict__ w2,
               const float* __restrict__ b2p, float* __restrict__ Araw,
               float* __restrict__ stats)
{
    const int wave = blockIdx.x * 8 + (threadIdx.x >> 5);
    const int lane = threadIdx.x & 31;
    const int i = wave & 127;            // receiver (column)
    const int j = (wave >> 7) & 127;     // sender   (row)
    const int b = wave >> 14;
    const float* ps = Ps + ((long)b * NREG + j) * NH_;
    const float* pr = Pr + ((long)b * NREG + i) * NH_;
    float acc = 0.f;
    if (i != j) {
        for (int c = lane; c < NH_; c += 32) {
            float v = ps[c] + pr[c] + b1v[c];
            v = act_elu(v);
            acc += v * w2[c];
        }
        for (int m = 16; m > 0; m >>= 1) acc += __shfl_xor(acc, m, 32);
    }
    __shared__ float ls[8], lq[8];
    const int w8 = threadIdx.x >> 5;
    if (lane == 0) {
        float xr = 0.f;
        if (i != j) xr = act_elu(acc + b2p[0]);
        Araw[wave] = xr;
        ls[w8] = (i != j) ? xr : 0.f;
        lq[w8] = (i != j) ? xr * xr : 0.f;
    }
    __syncthreads();
    if (threadIdx.x == 0) {
        float s = 0.f, q = 0.f;
        for (int t = 0; t < 8; ++t) { s += ls[t]; q += lq[t]; }
        atomicAdd(&stats[0], s);
        atomicAdd(&stats[1], q);
    }
}

// Apply f2's 1-feature batch-norm affine; zero the diagonal (non-edges).
__global__ __launch_bounds__(256)
void gate_finalize(const float* __restrict__ Araw, float* __restrict__ A,
                   const float* __restrict__ stats, const float* __restrict__ gp,
                   const float* __restrict__ bp)
{
    const long idx = (long)blockIdx.x * 256 + threadIdx.x;   // B*N*N
    const int i = idx & 127;
    const int j = (idx >> 7) & 127;
    const float cnt = (float)(BATCH_ * EDGES_);
    const float mu  = stats[0] / cnt;
    const float var = stats[1] / cnt - mu * mu;
    const float sc  = rsqrtf(var + 1e-5f) * gp[0];
    A[idx] = (i == j) ? 0.f : (Araw[idx] - mu) * sc + bp[0];
}

__global__ __launch_bounds__(256)
void concat_kernel(const float* __restrict__ x1, const float* __restrict__ z4,
                   float* __restrict__ out)
{
    const int idx = blockIdx.x * 256 + threadIdx.x;   // 1024*256
    const int r = idx >> 8, c = idx & 255;
    out[idx] = (c < NH_) ? x1[r * NH_ + c] : z4[r * NH_ + (c - NH_)];
}

// y1[b,f] = relu( z6flat[b]·W71[:,f] + b71[f] ).  grid = (128 f, 8 b), block 256.
__global__ __launch_bounds__(256)
void f71_kernel(const float* __restrict__ z6, const float* __restrict__ W,
                const float* __restrict__ bias, float* __restrict__ y1)
{
    const int f = blockIdx.x, b = blockIdx.y;
    const float* zr = z6 + (long)b * (NREG * NH_);
    float acc = 0.f;
    for (int k = threadIdx.x; k < NREG * NH_; k += 256)
        acc += zr[k] * W[(long)k * NH_ + f];
    __shared__ float red[256];
    red[threadIdx.x] = acc; __syncthreads();
    for (int st = 128; st > 0; st >>= 1) {
        if (threadIdx.x < st) red[threadIdx.x] += red[threadIdx.x + st];
        __syncthreads();
    }
    if (threadIdx.x == 0) y1[b * NH_ + f] = fmaxf(red[0] + bias[f], 0.f);
}

__global__ void f72_kernel(const float* __restrict__ y1, const float* __restrict__ W,
                           const float* __restrict__ bias, float* __restrict__ out)
{
    const int t = threadIdx.x;
    if (t < BATCH_ * 2) {
        const int b = t >> 1, o = t & 1;
        float acc = bias[o];
        for (int f = 0; f < NH_; ++f) acc += y1[b * NH_ + f] * W[f * 2 + o];
        out[b * 2 + o] = acc;
    }
}

__global__ void zero_kernel(float* p, int n)
{
    const int i = blockIdx.x * 256 + threadIdx.x;
    if (i < n) p[i] = 0.f;
}

// ---------------------------------------------------------------------------
// Input indices: top-level dict in insertion order; `params` pytree-flattened
// (dict keys sorted: f1,f2,f3,f4,f6,f71,f72,gat_W,gat_a; mlp: be,g,l1{b,w},l2{b,w}).
// ---------------------------------------------------------------------------
enum {
    IN_X = 0, IN_NODECLASS, IN_RECEIVE, IN_SEND, IN_TEMP,
    P_F1_BE, P_F1_G, P_F1_L1B, P_F1_L1W, P_F1_L2B, P_F1_L2W,
    P_F2_BE, P_F2_G, P_F2_L1B, P_F2_L1W, P_F2_L2B, P_F2_L2W,
    P_F3_BE, P_F3_G, P_F3_L1B, P_F3_L1W, P_F3_L2B, P_F3_L2W,
    P_F4_BE, P_F4_G, P_F4_L1B, P_F4_L1W, P_F4_L2B, P_F4_L2W,
    P_F6_BE, P_F6_G, P_F6_L1B, P_F6_L1W, P_F6_L2B, P_F6_L2W,
    P_F71_B, P_F71_W, P_F72_B, P_F72_W, P_GATW, P_GATA
};

extern "C" void kernel_launch(void* const* d_in, const int* in_sizes, int n_in,
                              void* d_out, int out_size, void* d_ws, size_t ws_size,
                              hipStream_t stream)
{
    auto F32 = [&](int i) { return (const float*)d_in[i]; };
    const float* x = F32(IN_X);                       // [B, N, NIN]

    float* ws = (float*)d_ws;
    float* stats   = ws;                              // 16
    float* x1      = ws + 16;                         // 131072
    float* x2      = x1 + ROWS_ * NH_;                // 131072
    float* gat_out = x2 + ROWS_ * NH_;                // 393216  [B][N][384]
    float* bufA    = gat_out + ROWS_ * (HEADS_*NH_);  // 524288
    float* bufB    = bufA + 524288;                   // 524288
    float* bufC    = bufB + 524288;                   // 262144
    float* y1      = bufC + 262144;                   // 1024

    auto gemm = [&](const float* A, const float* Bw, const float* bias, float* C,
                    int M, int K, int F, int lda, int ldb, int ldc,
                    long sA, long sB, long sC, int act, int batch) {
        dim3 g(F / 16, M / 64, batch);
        hipLaunchKernelGGL(gemm16_wmma, g, dim3(32), 0, stream,
                           A, Bw, bias, C, K, lda, ldb, ldc, sA, sB, sC, act);
    };

    hipLaunchKernelGGL(zero_kernel, dim3(1), dim3(256), 0, stream, stats, 16);

    // ---- f1 MLP: x -> x1 ------------------------------------------------
    gemm(x,    F32(P_F1_L1W), F32(P_F1_L1B), bufA, ROWS_, NIN_, NH_, NIN_, NH_, NH_, 0,0,0, ACT_ELU, 1);
    gemm(bufA, F32(P_F1_L2W), F32(P_F1_L2B), bufB, ROWS_, NH_,  NH_, NH_,  NH_, NH_, 0,0,0, ACT_ELU, 1);
    hipLaunchKernelGGL(bn_kernel, dim3(NH_), dim3(256), 0, stream,
                       bufB, x1, F32(P_F1_G), F32(P_F1_BE), ROWS_, NH_);

    // ---- f3 MLP: x -> x2 ------------------------------------------------
    gemm(x,    F32(P_F3_L1W), F32(P_F3_L1B), bufA, ROWS_, NIN_, NH_, NIN_, NH_, NH_, 0,0,0, ACT_ELU, 1);
    gemm(bufA, F32(P_F3_L2W), F32(P_F3_L2B), bufB, ROWS_, NH_,  NH_, NH_,  NH_, NH_, 0,0,0, ACT_ELU, 1);
    hipLaunchKernelGGL(bn_kernel, dim3(NH_), dim3(256), 0, stream,
                       bufB, x2, F32(P_F3_G), F32(P_F3_BE), ROWS_, NH_);

    // ---- GAT: x1 -> gat_out [B,N,3*NH] ---------------------------------
    float* Hbuf = bufA;                 // [B][3][N][NH]  (bh-contiguous)
    float* Att  = bufB;                 // [B][3][N][N]
    float* Sbuf = bufC;                 // [B*3][256]
    const long HB = (long)HEADS_ * NREG * NH_;   // per-batch stride in Hbuf/Att
    for (int hd = 0; hd < HEADS_; ++hd) {
        gemm(x1, F32(P_GATW) + (long)hd * NH_ * NH_, nullptr,
             Hbuf + (long)hd * NREG * NH_,
             NREG, NH_, NH_, NH_, NH_, NH_,
             (long)NREG * NH_, 0, HB, ACT_NONE, BATCH_);
    }
    hipLaunchKernelGGL(gat_scores, dim3(BATCH_ * HEADS_), dim3(128), 0, stream,
                       Hbuf, F32(P_GATA), Sbuf);
    hipLaunchKernelGGL(gat_att, dim3(BATCH_ * HEADS_ * NREG), dim3(128), 0, stream,
                       Sbuf, Att);
    for (int hd = 0; hd < HEADS_; ++hd) {
        gemm(Att  + (long)hd * NREG * NREG,
             Hbuf + (long)hd * NREG * NH_, nullptr,
             gat_out + hd * NH_,
             NREG, NREG, NH_, NREG, NH_, HEADS_ * NH_,
             HB, HB, (long)NREG * HEADS_ * NH_, ACT_ELU, BATCH_);
    }

    // ---- f2.l1 distributed: P_s = gat_out@W[0:384], P_r = gat_out@W[384:768]
    float* Ps = bufA;
    float* Pr = bufA + ROWS_ * NH_;
    gemm(gat_out, F32(P_F2_L1W),                   nullptr, Ps,
         ROWS_, HEADS_*NH_, NH_, HEADS_*NH_, NH_, NH_, 0,0,0, ACT_NONE, 1);
    gemm(gat_out, F32(P_F2_L1W) + HEADS_*NH_*NH_,  nullptr, Pr,
         ROWS_, HEADS_*NH_, NH_, HEADS_*NH_, NH_, NH_, 0,0,0, ACT_NONE, 1);

    // ---- per-edge gate + f2 batch-norm stats ---------------------------
    float* Araw  = bufB;
    float* Agate = bufB + BATCH_ * NREG * NREG;
    hipLaunchKernelGGL(edge_gate, dim3(BATCH_ * NREG * NREG / 8), dim3(256), 0, stream,
                       Ps, Pr, F32(P_F2_L1B), F32(P_F2_L2W), F32(P_F2_L2B), Araw, stats);
    hipLaunchKernelGGL(gate_finalize, dim3(BATCH_ * NREG * NREG / 256), dim3(256), 0, stream,
                       Araw, Agate, stats, F32(P_F2_G), F32(P_F2_BE));

    // ---- aggregate: z_node[b] = Agate[b] @ x2[b] -----------------------
    float* z_node = bufC;
    gemm(Agate, x2, nullptr, z_node,
         NREG, NREG, NH_, NREG, NH_, NH_,
         (long)NREG * NREG, (long)NREG * NH_, (long)NREG * NH_, ACT_NONE, BATCH_);

    // ---- f4 MLP: z_node -> z4 ------------------------------------------
    float* H4 = bufA;
    float* G4 = bufA + ROWS_ * NH_;
    float* z4 = bufB;
    gemm(z_node, F32(P_F4_L1W), F32(P_F4_L1B), H4, ROWS_, NH_, NH_, NH_, NH_, NH_, 0,0,0, ACT_ELU, 1);
    gemm(H4,     F32(P_F4_L2W), F32(P_F4_L2B), G4, ROWS_, NH_, NH_, NH_, NH_, NH_, 0,0,0, ACT_ELU, 1);
    hipLaunchKernelGGL(bn_kernel, dim3(NH_), dim3(256), 0, stream,
                       G4, z4, F32(P_F4_G), F32(P_F4_BE), ROWS_, NH_);

    // ---- f6 MLP on concat([x1, z4]) -> z6 ------------------------------
    float* Xcat = bufC;
    hipLaunchKernelGGL(concat_kernel, dim3(ROWS_ * 2 * NH_ / 256), dim3(256), 0, stream,
                       x1, z4, Xcat);
    float* H6 = bufA;
    float* G6 = bufA + ROWS_ * NH_;
    float* z6 = bufB;
    gemm(Xcat, F32(P_F6_L1W), F32(P_F6_L1B), H6, ROWS_, 2*NH_, NH_, 2*NH_, NH_, NH_, 0,0,0, ACT_ELU, 1);
    gemm(H6,   F32(P_F6_L2W), F32(P_F6_L2B), G6, ROWS_, NH_,   NH_, NH_,   NH_, NH_, 0,0,0, ACT_ELU, 1);
    hipLaunchKernelGGL(bn_kernel, dim3(NH_), dim3(256), 0, stream,
                       G6, z6, F32(P_F6_G), F32(P_F6_BE), ROWS_, NH_);

    // ---- heads: f71 (relu) then f72 ------------------------------------
    hipLaunchKernelGGL(f71_kernel, dim3(NH_, BATCH_), dim3(256), 0, stream,
                       z6, F32(P_F71_W), F32(P_F71_B), y1);
    hipLaunchKernelGGL(f72_kernel, dim3(1), dim3(32), 0, stream,
                       y1, F32(P_F72_W), F32(P_F72_B), (float*)d_out);
}